// SubgraphSpecificLatentProjector_47854525612388
// MI455X (gfx1250) — compile-verified
//
#include <hip/hip_runtime.h>
#include <hip/hip_bf16.h>
#include <math.h>

typedef __attribute__((ext_vector_type(2))) float v2f;
typedef __attribute__((ext_vector_type(8))) float v8f;

#define BB 256
#define NN 32
#define DD 128
#define RR 8
#define OO 8
#define NBUCKET 64
#define NCENTER (BB * RR)            // 2048 center rows
#define NROWS   (BB * RR + BB * NN)  // 10240 total rows
#define TROWS   32                   // rows per tile (2 WMMA row-subtiles)
#define MAXTILES 384                 // sum ceil(cnt/32) <= 320 + 63

// d_ws layout (int words)
#define WS_COUNTS  0
#define WS_OFFSETS 64
#define WS_CURSORS 128
#define WS_NTILES  192
#define WS_TILEB   256
#define WS_TILEO   (256 + 1024)
#define WS_ROWIDS  (256 + 2048)

__device__ __forceinline__ int row_key(int rid, const int* center_o,
                                       const int* s_types, const int* o_types) {
    if (rid < NCENTER) {
        int b = rid >> 3, r = rid & 7;
        int o = center_o[b]; o = o < 0 ? 0 : o;
        return r * OO + o;
    } else {
        int nid = rid - NCENTER;
        int s = s_types[nid]; s = s < 0 ? 0 : s;
        int o = o_types[nid]; o = o < 0 ? 0 : o;
        return s * OO + o;
    }
}

__global__ void k_zero(int* ws) {
    int t = threadIdx.x;
    if (t < NBUCKET) ws[WS_COUNTS + t] = 0;
    if (t == NBUCKET) ws[WS_NTILES] = 0;
}

__global__ void k_count(int* ws, const int* center_o, const int* s_types, const int* o_types) {
    int rid = blockIdx.x * blockDim.x + threadIdx.x;
    if (rid >= NROWS) return;
    atomicAdd(&ws[WS_COUNTS + row_key(rid, center_o, s_types, o_types)], 1);
}

__global__ void k_scan(int* ws) {
    __shared__ int sc[NBUCKET], st[NBUCKET];
    int t = threadIdx.x;
    int c = ws[WS_COUNTS + t];
    sc[t] = c;
    st[t] = (c + TROWS - 1) / TROWS;
    __syncthreads();
    if (t == 0) {
        int acc = 0;
        for (int i = 0; i < NBUCKET; ++i) { int v = sc[i]; sc[i] = acc; acc += v; }
        acc = 0;
        for (int i = 0; i < NBUCKET; ++i) { int v = st[i]; st[i] = acc; acc += v; }
        ws[WS_NTILES] = acc;
    }
    __syncthreads();
    ws[WS_OFFSETS + t] = sc[t];
    ws[WS_CURSORS + t] = sc[t];
    int ntb = (c + TROWS - 1) / TROWS;
    int base = st[t];
    for (int j = 0; j < ntb; ++j) {
        ws[WS_TILEB + base + j] = t;
        ws[WS_TILEO + base + j] = j * TROWS;
    }
}

__global__ void k_scatter(int* ws, const int* center_o, const int* s_types, const int* o_types) {
    int rid = blockIdx.x * blockDim.x + threadIdx.x;
    if (rid >= NROWS) return;
    int key = row_key(rid, center_o, s_types, o_types);
    int pos = atomicAdd(&ws[WS_CURSORS + key], 1);
    ws[WS_ROWIDS + pos] = rid;
}

// 384 blocks x 256 threads (8 wave32). One block = one tile of <=32 rows sharing
// one 128x128 matrix. Wave w owns output columns [16w, 16w+16) and computes TWO
// 16x16 row-subtiles, reusing each B fragment for two WMMAs (halves M traffic).
__global__ __launch_bounds__(256) void k_gemm(const int* __restrict__ ws,
                                              const float* __restrict__ center_h,
                                              const float* __restrict__ neigh,
                                              const float* __restrict__ M,
                                              const float* __restrict__ bias,
                                              float* __restrict__ out) {
    int tile = blockIdx.x;
    if (tile >= ws[WS_NTILES]) return;

    int key  = ws[WS_TILEB + tile];
    int toff = ws[WS_TILEO + tile];
    int cnt  = ws[WS_COUNTS + key];
    int off  = ws[WS_OFFSETS + key];

    // Stride 132 (mod 64 == 4) -> ds_load_b64 A-fragment reads are bank-conflict free.
    __shared__ float As[TROWS][132];
    __shared__ const float* s_src[TROWS];
    __shared__ int s_dst[TROWS];

    int t = threadIdx.x;
    if (t < TROWS) {
        int p = toff + t;
        if (p < cnt) {
            int rid = ws[WS_ROWIDS + off + p];
            if (rid < NCENTER) {
                int b = rid >> 3, r = rid & 7;
                s_src[t] = center_h + (size_t)b * DD;
                s_dst[t] = (b * RR + r) * DD;
            } else {
                int nid = rid - NCENTER;
                s_src[t] = neigh + (size_t)nid * DD;
                s_dst[t] = NCENTER * DD + nid * DD;
            }
        } else {
            s_src[t] = nullptr;
            s_dst[t] = -1;
        }
    }
    __syncthreads();

    // Cooperative stage of 32 x 128 activation rows into LDS (2 x 8 floats / thread).
    {
        int seg = t & 15;
#pragma unroll
        for (int h = 0; h < 2; ++h) {
            int row = (t >> 4) + h * 16;
            const float* sp = s_src[row];
            float4 v0 = make_float4(0.f, 0.f, 0.f, 0.f), v1 = v0;
            if (sp) {
                v0 = ((const float4*)sp)[seg * 2];
                v1 = ((const float4*)sp)[seg * 2 + 1];
            }
            *(float4*)&As[row][seg * 8]     = v0;
            *(float4*)&As[row][seg * 8 + 4] = v1;
        }
    }
    __syncthreads();

    int w    = t >> 5;
    int lane = t & 31;
    int nsub = lane & 15;          // A-row-in-subtile / B-column-in-subtile
    int koff = (lane >> 4) << 1;   // upper lane half supplies K+2, K+3
    int col  = w * 16 + nsub;

    const float* Mb    = M + (size_t)key * (DD * DD) + col; // column slice, row stride DD
    const float* Arow0 = &As[nsub][0];
    const float* Arow1 = &As[nsub + 16][0];

    v8f acc0 = {0.f, 0.f, 0.f, 0.f, 0.f, 0.f, 0.f, 0.f};
    v8f acc1 = acc0;

#pragma unroll 8
    for (int k = 0; k < DD; k += 4) {
        // Shared B fragment: lane half holds M[k+koff][col], M[k+koff+1][col]
        v2f bv; bv.x = Mb[(size_t)(k + koff) * DD];
                bv.y = Mb[(size_t)(k + koff + 1) * DD];
        // A fragments for the two row-subtiles
        float2 a0 = *(const float2*)&Arow0[k + koff];
        float2 a1 = *(const float2*)&Arow1[k + koff];
        v2f av0; av0.x = a0.x; av0.y = a0.y;
        v2f av1; av1.x = a1.x; av1.y = a1.y;
        acc0 = __builtin_amdgcn_wmma_f32_16x16x4_f32(false, av0, false, bv,
                                                     (short)0, acc0, false, false);
        acc1 = __builtin_amdgcn_wmma_f32_16x16x4_f32(false, av1, false, bv,
                                                     (short)0, acc1, false, false);
    }

    // Epilogue: C/D layout -> VGPR g, lanes 0-15: row g, lanes 16-31: row g+8.
    float bb    = bias[key * DD + col];
    int   rbase = (lane >> 4) << 3;
#pragma unroll
    for (int g = 0; g < 8; ++g) {
        int   row0 = rbase + g;
        int   d0   = s_dst[row0];
        float x0   = acc0[g] + bb;
        float y0   = 0.5f * x0 * (1.0f + erff(x0 * 0.70710678118654752f)); // exact GELU
        if (d0 >= 0) out[d0 + col] = y0;

        int   row1 = row0 + 16;
        int   d1   = s_dst[row1];
        float x1   = acc1[g] + bb;
        float y1   = 0.5f * x1 * (1.0f + erff(x1 * 0.70710678118654752f));
        if (d1 >= 0) out[d1 + col] = y1;
    }
}

extern "C" void kernel_launch(void* const* d_in, const int* in_sizes, int n_in,
                              void* d_out, int out_size, void* d_ws, size_t ws_size,
                              hipStream_t stream) {
    const float* center_h = (const float*)d_in[0];
    const float* neigh    = (const float*)d_in[1];
    const float* M        = (const float*)d_in[2];
    const float* bias     = (const float*)d_in[3];
    const int*   center_o = (const int*)d_in[4];
    const int*   s_types  = (const int*)d_in[5];
    const int*   o_types  = (const int*)d_in[6];
    float*       out      = (float*)d_out;
    int*         ws       = (int*)d_ws;

    k_zero<<<1, 128, 0, stream>>>(ws);
    k_count<<<(NROWS + 255) / 256, 256, 0, stream>>>(ws, center_o, s_types, o_types);
    k_scan<<<1, NBUCKET, 0, stream>>>(ws);
    k_scatter<<<(NROWS + 255) / 256, 256, 0, stream>>>(ws, center_o, s_types, o_types);
    k_gemm<<<MAXTILES, 256, 0, stream>>>(ws, center_h, neigh, M, bias, out);
}